// WB_29222957482419
// MI455X (gfx1250) — compile-verified
//
#include <hip/hip_runtime.h>
#include <hip/hip_bf16.h>

typedef __attribute__((ext_vector_type(2))) float v2f;
typedef __attribute__((ext_vector_type(4))) float v4f;
typedef __attribute__((ext_vector_type(8))) float v8f;

#define CC   16
#define HW   262144           // 512*512
#define CHW  (16*262144)
#define NSLICE 16
#define SLICE  (HW/NSLICE)    // 16384
#define FOLD_OFF 4096         // floats into ws
#define FOLD_STRIDE 544       // per-batch: A[256] avec[16] G[256] gvec[16]

// ---------------- Pass 1: per-(b,c) sum & max, sliced, deterministic, b128 loads ----------------
__global__ void wb_reduce_kernel(const float* __restrict__ x, float* __restrict__ ws) {
    __shared__ float ssum[256];
    __shared__ float smax[256];
    int bid   = blockIdx.x;        // ((b*16+c)*16 + slice)
    int slice = bid & 15;
    int bc    = bid >> 4;
    const v4f* p = (const v4f*)(x + (size_t)bc * HW + (size_t)slice * SLICE);
    float s = 0.0f, m = -3.402823466e38f;
    #pragma unroll 4
    for (int i = 0; i < SLICE/4/256; ++i) {      // 16 float4 per thread
        v4f v = p[i*256 + threadIdx.x];
        s += (v[0] + v[1]) + (v[2] + v[3]);
        m  = fmaxf(fmaxf(fmaxf(m, v[0]), fmaxf(v[1], v[2])), v[3]);
    }
    ssum[threadIdx.x] = s; smax[threadIdx.x] = m;
    __syncthreads();
    for (int off = 128; off > 0; off >>= 1) {
        if ((int)threadIdx.x < off) {
            ssum[threadIdx.x] += ssum[threadIdx.x + off];
            smax[threadIdx.x]  = fmaxf(smax[threadIdx.x], smax[threadIdx.x + off]);
        }
        __syncthreads();
    }
    if (threadIdx.x == 0) {
        ws[bid]        = ssum[0];   // partial sums: ws[0..2048)
        ws[2048 + bid] = smax[0];   // partial max : ws[2048..4096)
    }
}

// ---------------- Pass 2: fold broadcast divisions into 16x16 matrices ----------------
__global__ void wb_fold_kernel(const float* __restrict__ Wc, const float* __restrict__ bc,
                               const float* __restrict__ W1, const float* __restrict__ b1,
                               const float* __restrict__ W2, const float* __restrict__ b2,
                               const float* __restrict__ W3, const float* __restrict__ b3,
                               float* __restrict__ ws) {
    __shared__ float invm[16], invM[16];
    int b = blockIdx.x;
    int t = threadIdx.x;
    if (t < 16) {
        float s = 0.0f, m = -3.402823466e38f;
        int base = (b*16 + t) * 16;
        for (int i = 0; i < 16; ++i) {
            s += ws[base + i];
            m  = fmaxf(m, ws[2048 + base + i]);
        }
        invm[t] = 1.0f / (s * (1.0f/262144.0f) + 1e-5f);
        invM[t] = 1.0f / (m + 1e-5f);
    }
    __syncthreads();
    float* F = ws + FOLD_OFF + b*FOLD_STRIDE;
    int o = t >> 4, c = t & 15;
    float a = 0.0f, g = 0.0f;
    #pragma unroll
    for (int k = 0; k < 16; ++k) {
        a += W2[o*16 + k] * invm[k] * Wc[k*16 + c];
        g += W3[o*16 + k] * invM[k] * W1[k*16 + c];
    }
    F[t]       = a;   // A_b  [0,256)
    F[272 + t] = g;   // G_b  [272,528)
    if (t < 16) {
        float av = b2[t], gv = b3[t];
        #pragma unroll
        for (int k = 0; k < 16; ++k) {
            av += W2[t*16 + k] * invm[k] * bc[k];
            gv += W3[t*16 + k] * invM[k] * b1[k];
        }
        F[256 + t] = av;  // a_b [256,272)
        F[528 + t] = gv;  // g_b [528,544)
    }
}

// Returns 2*sigmoid(tanh(v)) - 1 = tanh(0.5*tanh(v))
__device__ __forceinline__ float wb_inner(float v) {
#if __has_builtin(__builtin_amdgcn_tanhf)
    return __builtin_amdgcn_tanhf(0.5f * __builtin_amdgcn_tanhf(v));
#else
    float e  = __builtin_amdgcn_exp2f(v * 2.8853900817779268f);   // e^(2v)
    float th = 1.0f - 2.0f * __builtin_amdgcn_rcpf(e + 1.0f);     // tanh(v)
    float e2 = __builtin_amdgcn_exp2f(th * 1.4426950408889634f);  // e^(tanh)
    return 1.0f - 2.0f * __builtin_amdgcn_rcpf(e2 + 1.0f);        // tanh(tanh/2)
#endif
}

// ---------------- Pass 3: WMMA main pass ----------------
// One wave = 16 tiles (4 groups of 4). Within a group of 64 pixels, tile t owns
// pixels pg + m*4 + t  (m = lane&15), so each lane's 4 tile-values per channel are
// 4 consecutive floats -> b128 loads/stores.
__global__ void __launch_bounds__(256)
wb_main_kernel(const float* __restrict__ x,
               const float* __restrict__ W4, const float* __restrict__ b4,
               const float* __restrict__ ws, float* __restrict__ out) {
    const int lane = threadIdx.x & 31;
    const int g    = lane >> 4;          // lane half-group
    const int m16  = lane & 15;
    const int wid  = blockIdx.x * 8 + (threadIdx.x >> 5);
    const long tile0 = (long)wid * 16;
    const int  b     = (int)(tile0 >> 14);           // 16384 tiles per batch
    const long p0    = (tile0 & 16383) << 4;         // pixel base within batch

    // Weight matrices in WMMA A-layout: lane m holds K = kk*4 + 2g + {0,1}
    const float* F = ws + FOLD_OFF + b*FOLD_STRIDE;
    const int koff = g*2;
    v2f Aw[4], Gw[4], Ww[4];
    #pragma unroll
    for (int kk = 0; kk < 4; ++kk) {
        int c0 = kk*4 + koff;
        Aw[kk][0] = F[m16*16 + c0];        Aw[kk][1] = F[m16*16 + c0 + 1];
        Gw[kk][0] = F[272 + m16*16 + c0];  Gw[kk][1] = F[272 + m16*16 + c0 + 1];
        Ww[kk][0] = W4[m16*16 + c0];       Ww[kk][1] = W4[m16*16 + c0 + 1];
    }
    // Biases in D-layout: reg r -> out-channel r + 8g
    v8f abias, gbias, obias;
    #pragma unroll
    for (int r = 0; r < 8; ++r) {
        abias[r] = F[256 + g*8 + r];
        gbias[r] = F[528 + g*8 + r];
        obias[r] = b4[g*8 + r];
    }

    const float* xb = x   + (size_t)b * CHW;
    float*       ob = out + (size_t)b * CHW;

    for (int gi = 0; gi < 4; ++gi) {
        const long pg = p0 + gi*64 + (long)m16*4;
        // 8 x b128: channel kk*4+2g+{0,1}, 4 consecutive pixels (one per tile)
        v4f Xq[4][2];
        #pragma unroll
        for (int kk = 0; kk < 4; ++kk) {
            int ch = kk*4 + koff;
            Xq[kk][0] = *(const v4f*)(xb + (size_t)ch     * HW + pg);
            Xq[kk][1] = *(const v4f*)(xb + (size_t)(ch+1) * HW + pg);
        }
        v4f Oq[8];
        #pragma unroll
        for (int t = 0; t < 4; ++t) {
            v2f Xb[4];
            #pragma unroll
            for (int kk = 0; kk < 4; ++kk) {
                Xb[kk][0] = Xq[kk][0][t];
                Xb[kk][1] = Xq[kk][1][t];
            }
            // wh = A_b X + a_b ; gray = G_b X + g_b
            v8f cw = abias, cg = gbias;
            #pragma unroll
            for (int kk = 0; kk < 4; ++kk) {
                cw = __builtin_amdgcn_wmma_f32_16x16x4_f32(false, Aw[kk], false, Xb[kk], (short)0, cw, false, false);
                cg = __builtin_amdgcn_wmma_f32_16x16x4_f32(false, Gw[kk], false, Xb[kk], (short)0, cg, false, false);
            }
            // s = 1 + sig(tanh(wh)) + sig(tanh(gray)) = 2 + 0.5*(inner(wh)+inner(gray))
            // lane owns channels 8g..8g+7 of its pixel
            float s_own[8], s_oth[8];
            #pragma unroll
            for (int r = 0; r < 8; ++r)
                s_own[r] = 2.0f + 0.5f * (wb_inner(cw[r]) + wb_inner(cg[r]));
            #pragma unroll
            for (int r = 0; r < 8; ++r)   // SWAPX16: exchange with lane^16
                s_oth[r] = __int_as_float(__builtin_amdgcn_ds_swizzle(__float_as_int(s_own[r]), 0x401F));
            // Scale X in B-layout: row K = kk*4 + 2g + j needs s[that channel]
            v2f Xs[4];
            #pragma unroll
            for (int kk = 0; kk < 4; ++kk) {
                #pragma unroll
                for (int j = 0; j < 2; ++j) {
                    int i0 = (kk & 1)*4 + j;
                    float sv = (kk < 2) ? (g ? s_oth[i0+2] : s_own[i0])
                                        : (g ? s_own[i0+2] : s_oth[i0]);
                    Xs[kk][j] = Xb[kk][j] * sv;
                }
            }
            // out = W4 (X .* s) + b4
            v8f co = obias;
            #pragma unroll
            for (int kk = 0; kk < 4; ++kk)
                co = __builtin_amdgcn_wmma_f32_16x16x4_f32(false, Ww[kk], false, Xs[kk], (short)0, co, false, false);
            #pragma unroll
            for (int r = 0; r < 8; ++r)
                Oq[r][t] = co[r];
        }
        // 8 x b128 streaming stores (don't evict x from L2)
        #pragma unroll
        for (int r = 0; r < 8; ++r)
            __builtin_nontemporal_store(Oq[r], (v4f*)(ob + (size_t)(g*8 + r) * HW + pg));
    }
}

extern "C" void kernel_launch(void* const* d_in, const int* in_sizes, int n_in,
                              void* d_out, int out_size, void* d_ws, size_t ws_size,
                              hipStream_t stream) {
    const float* x  = (const float*)d_in[0];
    const float* Wc = (const float*)d_in[1];
    const float* bc = (const float*)d_in[2];
    const float* W1 = (const float*)d_in[3];
    const float* b1 = (const float*)d_in[4];
    const float* W2 = (const float*)d_in[5];
    const float* b2 = (const float*)d_in[6];
    const float* W3 = (const float*)d_in[7];
    const float* b3 = (const float*)d_in[8];
    const float* W4 = (const float*)d_in[9];
    const float* b4 = (const float*)d_in[10];
    float* ws  = (float*)d_ws;
    float* out = (float*)d_out;

    // 8 batches * 16 chan * 16 slices = 2048 reduction blocks
    wb_reduce_kernel<<<2048, 256, 0, stream>>>(x, ws);
    wb_fold_kernel<<<8, 256, 0, stream>>>(Wc, bc, W1, b1, W2, b2, W3, b3, ws);
    // 131072 tiles / 16 per wave / 8 waves per block = 1024 blocks (exact, full waves)
    wb_main_kernel<<<1024, 256, 0, stream>>>(x, W4, b4, ws, out);
}